// SelfAttention3D_23450521436853
// MI455X (gfx1250) — compile-verified
//
#include <hip/hip_runtime.h>
#include <math.h>

// Problem constants (match reference)
#define BB 16
#define CC 512
#define NN 4096
#define HH 512

typedef __attribute__((ext_vector_type(2))) float v2f;
typedef __attribute__((ext_vector_type(4))) float v4f;
typedef __attribute__((ext_vector_type(8))) float v8f;

// ---------------------------------------------------------------------------
// xr[b,c] = sum_n x[b,c,n]   (one block per contiguous row, float4 loads)
// ---------------------------------------------------------------------------
__global__ void row_sum_kernel(const float* __restrict__ x, float* __restrict__ xr) {
    const int row = blockIdx.x;                       // b*CC + c
    const float* p = x + (size_t)row * NN;
    float s = 0.f;
    for (int i = threadIdx.x * 4; i < NN; i += blockDim.x * 4) {
        const float4 v = *(const float4*)(p + i);
        s += (v.x + v.y) + (v.z + v.w);
    }
    __shared__ float red[8];
    for (int o = 16; o > 0; o >>= 1) s += __shfl_xor(s, o, 32);
    const int wave = threadIdx.x >> 5, lane = threadIdx.x & 31;
    if (lane == 0) red[wave] = s;
    __syncthreads();
    if (threadIdx.x == 0) {
        float t = 0.f;
        for (int w = 0; w < (int)(blockDim.x >> 5); ++w) t += red[w];
        xr[row] = t;
    }
}

// ---------------------------------------------------------------------------
// xs[b,n] = sum_c x[b,c,n]   (thread per 4 n's, b128 loads coalesced)
// ---------------------------------------------------------------------------
__global__ void col_sum_kernel(const float* __restrict__ x, float* __restrict__ xs) {
    const int b  = blockIdx.y;
    const int n0 = (blockIdx.x * blockDim.x + threadIdx.x) * 4;
    const float* p = x + (size_t)b * CC * NN + n0;
    float4 s = {0.f, 0.f, 0.f, 0.f};
    for (int c = 0; c < CC; ++c) {
        const float4 v = *(const float4*)(p + (size_t)c * NN);
        s.x += v.x; s.y += v.y; s.z += v.z; s.w += v.w;
    }
    *(float4*)(xs + (size_t)b * NN + n0) = s;
}

// ---------------------------------------------------------------------------
// out[16 x ncols] = A[16 x K] * B   using V_WMMA_F32_16X16X4_F32.
//   TRANSB=1 : B[k,j] = W[j*K + k]     (W is [ncols,K] row-major, e.g. w_cf)
//   TRANSB=0 : B[k,j] = W[k*ncols + j] (W is [K,ncols] row-major, e.g. w_cg)
// One wave per 16-column tile; K-loop in steps of 4.
// A layout (ISA 7.12.2, 32-bit A 16x4): lane<16 -> K 0/1, lane>=16 -> K 2/3.
// C/D layout: VGPR r holds M=r (lanes 0-15) and M=8+r (lanes 16-31).
// ---------------------------------------------------------------------------
template <bool TRANSB>
__global__ void wmma_gemm16_kernel(const float* __restrict__ A,
                                   const float* __restrict__ W,
                                   float* __restrict__ out,
                                   int K, int ncols) {
    const int wave = threadIdx.x >> 5;
    const int lane = threadIdx.x & 31;
    const int tile = blockIdx.x * (blockDim.x >> 5) + wave;  // uniform per wave
    const int j0   = tile * 16;
    const int m    = lane & 15;            // A row / B column within tile
    const int ksel = (lane >> 4) << 1;     // 0 for lanes 0-15, 2 for lanes 16-31
    const int j    = j0 + m;

    v8f acc = {};
    for (int k0 = 0; k0 < K; k0 += 4) {
        v2f a, b;
        a.x = A[m * K + k0 + ksel];
        a.y = A[m * K + k0 + ksel + 1];
        if (TRANSB) {
            b.x = W[(size_t)j * K + k0 + ksel];
            b.y = W[(size_t)j * K + k0 + ksel + 1];
        } else {
            b.x = W[(size_t)(k0 + ksel)     * ncols + j];
            b.y = W[(size_t)(k0 + ksel + 1) * ncols + j];
        }
        // 8 args: (neg_a, A, neg_b, B, c_mod, C, reuse_a, reuse_b)
        acc = __builtin_amdgcn_wmma_f32_16x16x4_f32(
            false, a, false, b, (short)0, acc, false, false);
    }

    const int rbase = (lane >> 4) << 3;    // lanes 16-31 hold rows 8..15
#pragma unroll
    for (int r = 0; r < 8; ++r)
        out[(size_t)(rbase + r) * ncols + j] = acc[r];
}

// ---------------------------------------------------------------------------
// ch[b,c] = dot(x[b,c,:], v[b,:])   (one block per row, float4 loads)
// ---------------------------------------------------------------------------
__global__ void chcomp_kernel(const float* __restrict__ x, const float* __restrict__ v,
                              float* __restrict__ ch) {
    const int row = blockIdx.x;                 // b*CC + c
    const int b = row / CC;
    const float* px = x + (size_t)row * NN;
    const float* pv = v + (size_t)b * NN;
    float s = 0.f;
    for (int i = threadIdx.x * 4; i < NN; i += blockDim.x * 4) {
        const float4 a = *(const float4*)(px + i);
        const float4 w = *(const float4*)(pv + i);
        s += a.x * w.x + a.y * w.y + a.z * w.z + a.w * w.w;
    }
    __shared__ float red[8];
    for (int o = 16; o > 0; o >>= 1) s += __shfl_xor(s, o, 32);
    const int wave = threadIdx.x >> 5, lane = threadIdx.x & 31;
    if (lane == 0) red[wave] = s;
    __syncthreads();
    if (threadIdx.x == 0) {
        float t = 0.f;
        for (int w = 0; w < (int)(blockDim.x >> 5); ++w) t += red[w];
        ch[row] = t;
    }
}

// ---------------------------------------------------------------------------
// sp[b,n] = sum_c x[b,c,n] * q[b,c]   (thread per 4 n's, b128 loads)
// ---------------------------------------------------------------------------
__global__ void spcomp_kernel(const float* __restrict__ x, const float* __restrict__ q,
                              float* __restrict__ sp) {
    const int b  = blockIdx.y;
    const int n0 = (blockIdx.x * blockDim.x + threadIdx.x) * 4;
    const float* px = x + (size_t)b * CC * NN + n0;
    const float* pq = q + b * CC;
    float4 s = {0.f, 0.f, 0.f, 0.f};
    for (int c = 0; c < CC; ++c) {
        const float  qc = pq[c];
        const float4 v  = *(const float4*)(px + (size_t)c * NN);
        s.x += v.x * qc; s.y += v.y * qc; s.z += v.z * qc; s.w += v.w * qc;
    }
    *(float4*)(sp + (size_t)b * NN + n0) = s;
}

// ---------------------------------------------------------------------------
// softmax over `len` elements per block row (standard max-sub softmax)
// ---------------------------------------------------------------------------
__global__ void softmax_kernel(const float* __restrict__ in, float* __restrict__ out, int len) {
    const int b = blockIdx.x;
    const float* p = in + (size_t)b * len;
    const int wave = threadIdx.x >> 5, lane = threadIdx.x & 31;
    const int nwaves = blockDim.x >> 5;
    __shared__ float redm[8];
    __shared__ float reds[8];

    float m = -INFINITY;
    for (int i = threadIdx.x; i < len; i += blockDim.x) m = fmaxf(m, p[i]);
    for (int o = 16; o > 0; o >>= 1) m = fmaxf(m, __shfl_xor(m, o, 32));
    if (lane == 0) redm[wave] = m;
    __syncthreads();
    m = redm[0];
    for (int w = 1; w < nwaves; ++w) m = fmaxf(m, redm[w]);

    float s = 0.f;
    for (int i = threadIdx.x; i < len; i += blockDim.x) s += expf(p[i] - m);
    for (int o = 16; o > 0; o >>= 1) s += __shfl_xor(s, o, 32);
    if (lane == 0) reds[wave] = s;
    __syncthreads();
    s = 0.f;
    for (int w = 0; w < nwaves; ++w) s += reds[w];
    const float inv = 1.f / s;

    for (int i = threadIdx.x; i < len; i += blockDim.x)
        out[(size_t)b * len + i] = expf(p[i] - m) * inv;
}

// ---------------------------------------------------------------------------
// out[b,c,n] = x[b,c,n] * cm[b,c] * sm[b,n]   (float4, non-temporal stores so
// x stays L2-resident across graph replays: x = 134 MB < 192 MB global L2)
// ---------------------------------------------------------------------------
__global__ void finalize_kernel(const float* __restrict__ x, const float* __restrict__ cm,
                                const float* __restrict__ sm, float* __restrict__ out) {
    const size_t base = ((size_t)blockIdx.x * blockDim.x + threadIdx.x) * 4;
    const int n  = (int)(base % NN);          // NN multiple of 4 -> same (b,c) for all 4
    const int bc = (int)(base / NN);
    const int b  = bc / CC;
    const float cmv = cm[bc];
    const v4f xv = *(const v4f*)(x + base);
    const v4f sv = *(const v4f*)(sm + (size_t)b * NN + n);
    v4f o = xv * cmv * sv;
    __builtin_nontemporal_store(o, (v4f*)(out + base));
}

// ---------------------------------------------------------------------------
extern "C" void kernel_launch(void* const* d_in, const int* in_sizes, int n_in,
                              void* d_out, int out_size, void* d_ws, size_t ws_size,
                              hipStream_t stream) {
    const float* x    = (const float*)d_in[0];   // [B, C, N]
    const float* w_cf = (const float*)d_in[1];   // [H, N]
    const float* w_cg = (const float*)d_in[2];   // [H, N]
    const float* w_sf = (const float*)d_in[3];   // [H, C]
    const float* w_sg = (const float*)d_in[4];   // [H, C]
    float* out = (float*)d_out;

    // d_out layout: outputs [B*C*N] | channel_mask [B*C] | spatial_mask [B*N]
    float* cm_out = out + (size_t)BB * CC * NN;
    float* sm_out = cm_out + (size_t)BB * CC;

    // workspace carve-up (floats)
    float* ws = (float*)d_ws;
    float* xr = ws;                 ws += BB * CC;   // [B,C]  row sums
    float* xs = ws;                 ws += BB * NN;   // [B,N]  col sums
    float* u  = ws;                 ws += BB * HH;   // [B,H]
    float* p  = ws;                 ws += BB * HH;   // [B,H]
    float* v  = ws;                 ws += BB * NN;   // [B,N]
    float* q  = ws;                 ws += BB * CC;   // [B,C]
    float* ch = ws;                 ws += BB * CC;   // [B,C]  pre-softmax
    float* sp = ws;                 /* ws += BB*NN */ // [B,N] pre-softmax

    // 1) reductions over x (two streaming passes, float4)
    row_sum_kernel<<<BB * CC, 256, 0, stream>>>(x, xr);
    col_sum_kernel<<<dim3(NN / (256 * 4), BB), 256, 0, stream>>>(x, xs);

    // 2) tiny GEMMs on WMMA f32 16x16x4 (4 waves/block, 1 tile/wave)
    //    u[b,h] = xs[b,:] . w_cf[h,:]     (B transposed read)
    wmma_gemm16_kernel<true ><<<(HH / 16) / 4, 128, 0, stream>>>(xs, w_cf, u, NN, HH);
    //    p[b,h] = xr[b,:] . w_sf[h,:]
    wmma_gemm16_kernel<true ><<<(HH / 16) / 4, 128, 0, stream>>>(xr, w_sf, p, CC, HH);
    //    v[b,n] = sum_h u[b,h] w_cg[h,n]  (B natural read)
    wmma_gemm16_kernel<false><<<(NN / 16) / 4, 128, 0, stream>>>(u, w_cg, v, HH, NN);
    //    q[b,c] = sum_h p[b,h] w_sg[h,c]
    wmma_gemm16_kernel<false><<<(CC / 16) / 4, 128, 0, stream>>>(p, w_sg, q, HH, CC);

    // 3) pre-softmax logits (two more streaming passes over x, float4)
    chcomp_kernel<<<BB * CC, 256, 0, stream>>>(x, v, ch);
    spcomp_kernel<<<dim3(NN / (256 * 4), BB), 256, 0, stream>>>(x, q, sp);

    // 4) softmaxes, written directly into the d_out mask region
    softmax_kernel<<<BB, 256, 0, stream>>>(ch, cm_out, CC);
    softmax_kernel<<<BB, 256, 0, stream>>>(sp, sm_out, NN);

    // 5) elementwise finalize: out = x * cm[b,c] * sm[b,n]
    const size_t total = (size_t)BB * CC * NN;
    finalize_kernel<<<(unsigned)(total / (256 * 4)), 256, 0, stream>>>(x, cm_out, sm_out, out);
}